// GIN_Classic_31482110280433
// MI455X (gfx1250) — compile-verified
//
#include <hip/hip_runtime.h>
#include <hip/hip_bf16.h>

typedef __attribute__((ext_vector_type(2))) float v2f;
typedef __attribute__((ext_vector_type(8))) float v8f;

#define N_NODES 50000
#define N_EDGES 800000
#define HDIM 128
#define NLAYERS 4
#define NGRAPHS 256
#define OUTDIM 10
#define POOLW (NLAYERS * HDIM)   // 512
#define BN_EPS 1e-5f
#define SCAN_CHUNK 512
#define NCHUNKS ((N_NODES + SCAN_CHUNK - 1) / SCAN_CHUNK)   // 98

// ===========================================================================
// CSR construction (edge_index is layer-invariant: build once per launch)
// ===========================================================================
__global__ void count_deg(const int* __restrict__ ei, int* __restrict__ counts) {
    int e = blockIdx.x * blockDim.x + threadIdx.x;
    if (e < N_EDGES) atomicAdd(&counts[ei[N_EDGES + e]], 1);
}

__global__ void chunk_sum(const int* __restrict__ counts, int* __restrict__ chunkSums) {
    __shared__ int sm[SCAN_CHUNK];
    int t = threadIdx.x;
    int i = blockIdx.x * SCAN_CHUNK + t;
    sm[t] = (i < N_NODES) ? counts[i] : 0;
    __syncthreads();
    for (int s = SCAN_CHUNK / 2; s > 0; s >>= 1) {
        if (t < s) sm[t] += sm[t + s];
        __syncthreads();
    }
    if (t == 0) chunkSums[blockIdx.x] = sm[0];
}

__global__ void scan_chunks(int* __restrict__ chunkSums, int n) {
    if (blockIdx.x == 0 && threadIdx.x == 0) {
        int acc = 0;
        for (int i = 0; i < n; ++i) {
            int v = chunkSums[i];
            chunkSums[i] = acc;
            acc += v;
        }
    }
}

__global__ void chunk_scan_write(const int* __restrict__ counts,
                                 const int* __restrict__ chunkSums,
                                 int* __restrict__ rowoff, int* __restrict__ cursor) {
    __shared__ int sm[SCAN_CHUNK];
    int t = threadIdx.x;
    int i = blockIdx.x * SCAN_CHUNK + t;
    int v = (i < N_NODES) ? counts[i] : 0;
    sm[t] = v;
    __syncthreads();
    // inclusive Hillis-Steele scan
    for (int s = 1; s < SCAN_CHUNK; s <<= 1) {
        int add = (t >= s) ? sm[t - s] : 0;
        __syncthreads();
        sm[t] += add;
        __syncthreads();
    }
    int excl = sm[t] - v + chunkSums[blockIdx.x];
    if (i < N_NODES) {
        rowoff[i] = excl;
        cursor[i] = excl;
    }
    if (i == N_NODES - 1) rowoff[N_NODES] = excl + v;
}

__global__ void fill_csr(const int* __restrict__ ei, int* __restrict__ cursor,
                         int* __restrict__ csr_src) {
    int e = blockIdx.x * blockDim.x + threadIdx.x;
    if (e < N_EDGES) {
        int dst = ei[N_EDGES + e];
        int pos = atomicAdd(&cursor[dst], 1);
        csr_src[pos] = ei[e];
    }
}

// ===========================================================================
// agg[i] = x[i] + sum_{j in N(i)} x[j]   (pure gather, no atomics)
// one wave per node: 32 lanes x float4 = 128 features
// ===========================================================================
__global__ void gather_agg(const float* __restrict__ x, const int* __restrict__ rowoff,
                           const int* __restrict__ csr_src, float* __restrict__ agg) {
    int node = blockIdx.x * (blockDim.x / 32) + (threadIdx.x >> 5);
    int lane = threadIdx.x & 31;
    if (node >= N_NODES) return;
    int c = lane * 4;
    float4 acc = *(const float4*)(x + (size_t)node * HDIM + c);
    int beg = rowoff[node], end = rowoff[node + 1];
    for (int e = beg; e < end; ++e) {
        int s = csr_src[e];
        const float4 v = *(const float4*)(x + (size_t)s * HDIM + c);
        acc.x += v.x; acc.y += v.y; acc.z += v.z; acc.w += v.w;
    }
    *(float4*)(agg + (size_t)node * HDIM + c) = acc;
}

// ===========================================================================
// h_pre = A @ W1 + b1   (fp32 WMMA 16x16x4, 16-row tile per wave)
// also accumulates per-feature sum / sumsq for BatchNorm.
// ===========================================================================
__global__ void __launch_bounds__(32)
gemm1_wmma(const float* __restrict__ A, const float* __restrict__ W,
           const float* __restrict__ bias, float* __restrict__ hpre,
           float* __restrict__ colsum, float* __restrict__ colsumsq) {
    const int lane = threadIdx.x;
    const int half = lane >> 4;      // 0: K 0..1, rows M ; 1: K 2..3, rows M+8
    const int l15  = lane & 15;
    const int row0 = blockIdx.x * 16;

    const v8f vzero = {0.f, 0.f, 0.f, 0.f, 0.f, 0.f, 0.f, 0.f};
    v8f acc[8];
#pragma unroll
    for (int nt = 0; nt < 8; ++nt) acc[nt] = vzero;

    const float* arow = A + (size_t)(row0 + l15) * HDIM + half * 2;
    for (int k0 = 0; k0 < HDIM; k0 += 4) {
        v2f a;
        a.x = arow[k0];
        a.y = arow[k0 + 1];
        const float* wk = W + (size_t)(k0 + half * 2) * HDIM + l15;
#pragma unroll
        for (int nt = 0; nt < 8; ++nt) {
            v2f b;
            b.x = wk[nt * 16];
            b.y = wk[HDIM + nt * 16];
            acc[nt] = __builtin_amdgcn_wmma_f32_16x16x4_f32(
                false, a, false, b, (short)0, acc[nt], false, false);
        }
    }

#pragma unroll
    for (int nt = 0; nt < 8; ++nt) {
        int n = nt * 16 + l15;
        float bv = bias[n];
        float s = 0.f, sq = 0.f;
#pragma unroll
        for (int r = 0; r < 8; ++r) {
            float v = acc[nt][r] + bv;
            int row = row0 + half * 8 + r;
            hpre[(size_t)row * HDIM + n] = v;
            s += v;
            sq += v * v;
        }
        atomicAdd(&colsum[n], s);
        atomicAdd(&colsumsq[n], sq);
    }
}

// ===========================================================================
// Fold BN stats into per-feature scale/shift.
// ===========================================================================
__global__ void bn_finalize(const float* __restrict__ colsum,
                            const float* __restrict__ colsumsq,
                            const float* __restrict__ gamma,
                            const float* __restrict__ beta,
                            float* __restrict__ scale, float* __restrict__ shift,
                            float inv_n) {
    int t = threadIdx.x;   // 128 threads
    float mean = colsum[t] * inv_n;
    float var  = colsumsq[t] * inv_n - mean * mean;
    float sc   = gamma[t] * rsqrtf(var + BN_EPS);
    scale[t] = sc;
    shift[t] = beta[t] - mean * sc;
}

// ===========================================================================
// x_out = relu(BN(hpre)) @ W2 + b2 ; pooled[batch[row], l*H + n] += x_out
// (run-length compressed pooled atomics -- batch is sorted)
// ===========================================================================
__global__ void __launch_bounds__(32)
gemm2_wmma(const float* __restrict__ hpre, const float* __restrict__ W,
           const float* __restrict__ bias, const float* __restrict__ scale,
           const float* __restrict__ shift, const int* __restrict__ batch,
           float* __restrict__ xout, float* __restrict__ pooled, int layer) {
    const int lane = threadIdx.x;
    const int half = lane >> 4;
    const int l15  = lane & 15;
    const int row0 = blockIdx.x * 16;

    const v8f vzero = {0.f, 0.f, 0.f, 0.f, 0.f, 0.f, 0.f, 0.f};
    v8f acc[8];
#pragma unroll
    for (int nt = 0; nt < 8; ++nt) acc[nt] = vzero;

    const float* arow = hpre + (size_t)(row0 + l15) * HDIM + half * 2;
    for (int k0 = 0; k0 < HDIM; k0 += 4) {
        int k = k0 + half * 2;
        float a0 = fmaxf(arow[k0]     * scale[k]     + shift[k],     0.f);
        float a1 = fmaxf(arow[k0 + 1] * scale[k + 1] + shift[k + 1], 0.f);
        v2f a; a.x = a0; a.y = a1;
        const float* wk = W + (size_t)k * HDIM + l15;
#pragma unroll
        for (int nt = 0; nt < 8; ++nt) {
            v2f b;
            b.x = wk[nt * 16];
            b.y = wk[HDIM + nt * 16];
            acc[nt] = __builtin_amdgcn_wmma_f32_16x16x4_f32(
                false, a, false, b, (short)0, acc[nt], false, false);
        }
    }

    // rows this lane owns (same for all column tiles); batch is sorted
    int gidx[8];
#pragma unroll
    for (int r = 0; r < 8; ++r) gidx[r] = batch[row0 + half * 8 + r];

#pragma unroll
    for (int nt = 0; nt < 8; ++nt) {
        int n = nt * 16 + l15;
        float bv = bias[n];
        int curg = gidx[0];
        float psum = 0.f;
#pragma unroll
        for (int r = 0; r < 8; ++r) {
            float v = acc[nt][r] + bv;
            int row = row0 + half * 8 + r;
            xout[(size_t)row * HDIM + n] = v;
            if (gidx[r] != curg) {
                atomicAdd(&pooled[(size_t)curg * POOLW + layer * HDIM + n], psum);
                curg = gidx[r];
                psum = 0.f;
            }
            psum += v;
        }
        atomicAdd(&pooled[(size_t)curg * POOLW + layer * HDIM + n], psum);
    }
}

// ===========================================================================
// z1 = pooled @ Wp1 + bp1 ; accumulate BN stats over G
// ===========================================================================
__global__ void pool_fc1(const float* __restrict__ z, const float* __restrict__ Wp1,
                         const float* __restrict__ bp1, float* __restrict__ z1,
                         float* __restrict__ csum, float* __restrict__ csq) {
    __shared__ float zr[POOLW];
    int g = blockIdx.x, t = threadIdx.x;   // 128 threads
#pragma unroll
    for (int j = 0; j < POOLW / 128; ++j)
        zr[t + j * 128] = z[(size_t)g * POOLW + t + j * 128];
    __syncthreads();
    float s = bp1[t];
    for (int k = 0; k < POOLW; ++k)
        s += zr[k] * Wp1[(size_t)k * HDIM + t];
    z1[(size_t)g * HDIM + t] = s;
    atomicAdd(&csum[t], s);
    atomicAdd(&csq[t], s * s);
}

// ===========================================================================
// out = relu(BN(z1)) @ Wp2 + bp2
// ===========================================================================
__global__ void pool_fc2(const float* __restrict__ z1, const float* __restrict__ scale,
                         const float* __restrict__ shift, const float* __restrict__ Wp2,
                         const float* __restrict__ bp2, float* __restrict__ out) {
    __shared__ float av[HDIM];
    int g = blockIdx.x, t = threadIdx.x;   // 128 threads
    float a = z1[(size_t)g * HDIM + t] * scale[t] + shift[t];
    av[t] = fmaxf(a, 0.f);
    __syncthreads();
    if (t < OUTDIM) {
        float s = bp2[t];
        for (int k = 0; k < HDIM; ++k)
            s += av[k] * Wp2[k * OUTDIM + t];
        out[g * OUTDIM + t] = s;
    }
}

// ===========================================================================
extern "C" void kernel_launch(void* const* d_in, const int* in_sizes, int n_in,
                              void* d_out, int out_size, void* d_ws, size_t ws_size,
                              hipStream_t stream) {
    const float* x     = (const float*)d_in[0];
    const int*   ei    = (const int*)d_in[1];
    const int*   batch = (const int*)d_in[2];
    const float* W1    = (const float*)d_in[4];
    const float* b1    = (const float*)d_in[5];
    const float* gamma = (const float*)d_in[6];
    const float* beta  = (const float*)d_in[7];
    const float* W2    = (const float*)d_in[8];
    const float* b2    = (const float*)d_in[9];
    const float* Wp1   = (const float*)d_in[10];
    const float* bp1   = (const float*)d_in[11];
    const float* gp    = (const float*)d_in[12];
    const float* bp    = (const float*)d_in[13];
    const float* Wp2   = (const float*)d_in[14];
    const float* bp2   = (const float*)d_in[15];

    const size_t BIG = (size_t)N_NODES * HDIM * sizeof(float);   // 25,600,000 B
    char* ws = (char*)d_ws;
    float* agg    = (float*)(ws);
    float* hpre   = (float*)(ws + BIG);
    float* xbuf   = (float*)(ws + 2 * BIG);
    float* pooled = (float*)(ws + 3 * BIG);                       // G*512 floats
    float* statsL = (float*)(ws + 3 * BIG + (size_t)NGRAPHS * POOLW * sizeof(float));
    float* pstats = statsL + NLAYERS * 256;   // [2][128]
    float* scl    = pstats + 256;
    float* shf    = scl + 128;
    float* pscl   = shf + 128;
    float* pshf   = pscl + 128;
    float* z1     = pshf + 128;               // G*128 floats
    int* counts    = (int*)(z1 + (size_t)NGRAPHS * HDIM);
    int* rowoff    = counts + N_NODES;        // N+1 entries
    int* cursor    = rowoff + N_NODES + 1;
    int* chunkSums = cursor + N_NODES;        // NCHUNKS entries (<=128)
    int* csr_src   = chunkSums + 128;         // E entries

    // zero pooled + stat accumulators + degree counts (once, graph-capture safe)
    hipMemsetAsync(pooled, 0,
                   (size_t)NGRAPHS * POOLW * sizeof(float) +
                   (NLAYERS * 256 + 256) * sizeof(float), stream);
    hipMemsetAsync(counts, 0, (size_t)N_NODES * sizeof(int), stream);

    // ---- build CSR once (edge_index constant across layers) ----
    count_deg<<<(N_EDGES + 255) / 256, 256, 0, stream>>>(ei, counts);
    chunk_sum<<<NCHUNKS, SCAN_CHUNK, 0, stream>>>(counts, chunkSums);
    scan_chunks<<<1, 1, 0, stream>>>(chunkSums, NCHUNKS);
    chunk_scan_write<<<NCHUNKS, SCAN_CHUNK, 0, stream>>>(counts, chunkSums,
                                                         rowoff, cursor);
    fill_csr<<<(N_EDGES + 255) / 256, 256, 0, stream>>>(ei, cursor, csr_src);

    // ---- GIN layers ----
    const float* xin = x;
    for (int l = 0; l < NLAYERS; ++l) {
        float* csum = statsL + l * 256;
        float* csq  = csum + 128;

        gather_agg<<<(N_NODES + 7) / 8, 256, 0, stream>>>(xin, rowoff, csr_src, agg);
        gemm1_wmma<<<N_NODES / 16, 32, 0, stream>>>(
            agg, W1 + (size_t)l * HDIM * HDIM, b1 + l * HDIM, hpre, csum, csq);
        bn_finalize<<<1, 128, 0, stream>>>(
            csum, csq, gamma + l * HDIM, beta + l * HDIM, scl, shf, 1.0f / N_NODES);
        gemm2_wmma<<<N_NODES / 16, 32, 0, stream>>>(
            hpre, W2 + (size_t)l * HDIM * HDIM, b2 + l * HDIM, scl, shf, batch,
            xbuf, pooled, l);
        xin = xbuf;
    }

    // ---- pooled MLP head ----
    float* pcsum = pstats;
    float* pcsq  = pstats + 128;
    pool_fc1<<<NGRAPHS, 128, 0, stream>>>(pooled, Wp1, bp1, z1, pcsum, pcsq);
    bn_finalize<<<1, 128, 0, stream>>>(pcsum, pcsq, gp, bp, pscl, pshf,
                                       1.0f / NGRAPHS);
    pool_fc2<<<NGRAPHS, 128, 0, stream>>>(z1, pscl, pshf, Wp2, bp2, (float*)d_out);
}